// PlackettLuceLoss_4294967296158
// MI455X (gfx1250) — compile-verified
//
#include <hip/hip_runtime.h>
#include <hip/hip_bf16.h>

#define T_LEN 8192
#define BLOCK 1024
#define SEG   8          // labels / elements per thread (8192 / 1024)

// ---- LDS carve (dynamic shared) ----
// vexp  : 32768 B  @ 0      (float[8192])  -- TDM DMA target, then exp() in place
// lab   : 32768 B  @ 32768  (int  [8192])  -- TDM DMA target (labels row)
// cur   : 32768 B  @ 65536  (uint [8192])  -- histogram counts, then scatter cursors
// sidx  : 16384 B  @ 98304  (u16  [8192])  -- bucket-scattered original indices
// redF  :   132 B  @ 114688 (float[33])
// redU  :   132 B  @ 114824 (uint [33])
#define SMEM_BYTES 114960

typedef unsigned int u32x4_t __attribute__((ext_vector_type(4)));
typedef int          i32x4_t __attribute__((ext_vector_type(4)));
typedef int          i32x8_t __attribute__((ext_vector_type(8)));

__device__ __forceinline__ unsigned waveInclScanU(unsigned v, int lane) {
#pragma unroll
  for (int d = 1; d < 32; d <<= 1) {
    unsigned n = __shfl_up(v, d, 32);
    if (lane >= d) v += n;
  }
  return v;
}

__device__ __forceinline__ float waveInclScanF(float v, int lane) {
#pragma unroll
  for (int d = 1; d < 32; d <<= 1) {
    float n = __shfl_up(v, d, 32);
    if (lane >= d) v += n;
  }
  return v;
}

#if defined(__HIP_DEVICE_COMPILE__)
// Issue one TDM descriptor: 1-D row of T_LEN 4-byte elements, global -> LDS.
__device__ __forceinline__ void tdm_load_row_4B(const void* gptr, unsigned ldsOff) {
  unsigned long long ga = (unsigned long long)gptr;
  u32x4_t g0;
  g0[0] = 1u;                                                  // count = 1 descriptor
  g0[1] = ldsOff;                                              // lds_addr (bytes)
  g0[2] = (unsigned)(ga & 0xFFFFFFFFull);                      // global_addr[31:0]
  g0[3] = (unsigned)((ga >> 32) & 0x01FFFFFFull) | (2u << 30); // addr[56:32] | type=2
  i32x8_t g1;
  g1[0] = 0x00020000;                 // workgroup_mask=0, data_size=2 (4 bytes)
  g1[1] = (int)(0x2000u << 16);       // tensor_dim0 = 8192 (low 16 at bits 63:48)
  g1[2] = 0x00010000;                 // tensor_dim0 hi16 = 0, tensor_dim1 = 1
  g1[3] = 0x20000000;                 // tile_dim0 = 8192 (bits 127:112)
  g1[4] = 0x00000001;                 // tile_dim1 = 1, tile_dim2 = 0
  g1[5] = 0x00002000;                 // tensor_dim0_stride = 8192 (lo32)
  g1[6] = 0;
  g1[7] = 0;
  i32x4_t gz4 = {0, 0, 0, 0};
#if __has_include(<hip/amd_detail/amd_gfx1250_TDM.h>)
  i32x8_t gz8 = {0, 0, 0, 0, 0, 0, 0, 0};
  __builtin_amdgcn_tensor_load_to_lds(g0, g1, gz4, gz4, gz8, 0);
#else
  __builtin_amdgcn_tensor_load_to_lds(g0, g1, gz4, gz4, 0);
#endif
}
#endif

__global__ __launch_bounds__(BLOCK) void pl_row_kernel(const float* __restrict__ logits,
                                                       const int* __restrict__ labels,
                                                       float* __restrict__ row_out) {
  extern __shared__ unsigned char smem[];
  float*          vexp = (float*)smem;                     // offset 0      (TDM dest)
  int*            lab  = (int*)(smem + 32768);             // offset 32768  (TDM dest)
  unsigned*       cur  = (unsigned*)(smem + 65536);
  unsigned short* sidx = (unsigned short*)(smem + 98304);
  float*          redF = (float*)(smem + 114688);          // 33 floats
  unsigned*       redU = (unsigned*)(smem + 114824);       // 33 uints

  const int    tid     = threadIdx.x;
  const int    lane    = tid & 31;
  const int    wid     = tid >> 5;
  const int    row     = blockIdx.x;
  const size_t rowBase = (size_t)row * T_LEN;

  // zero per-label counters
#pragma unroll
  for (int k = 0; k < SEG; ++k) cur[tid + k * BLOCK] = 0u;

  // ---- P0: TDM async DMA of logits row + labels row (32KB each) into LDS ----
#if defined(__HIP_DEVICE_COMPILE__)
  if (wid == 0) {
    unsigned base = (unsigned)(unsigned long long)(size_t)smem; // LDS byte offset
    tdm_load_row_4B((const void*)(logits + rowBase), base);
    tdm_load_row_4B((const void*)(labels + rowBase), base + 32768u);
    __builtin_amdgcn_s_wait_tensorcnt(0);
  }
#endif
  __syncthreads();   // rows resident in LDS, counters zeroed

  // ---- P1: exp() in place, label histogram, running score sum ----
  float ssum = 0.0f;
#pragma unroll
  for (int k = 0; k < SEG; ++k) {
    int i = tid + k * BLOCK;
    float s = vexp[i];
    ssum += s;
    vexp[i] = expf(s);
    atomicAdd(&cur[lab[i]], 1u);
  }
  __syncthreads();

  // ---- P2: block exclusive scan of counts -> bucket starts / cursors ----
  const int L0 = tid * SEG;
  unsigned cnt8[SEG];
  unsigned tsum = 0;
#pragma unroll
  for (int k = 0; k < SEG; ++k) { cnt8[k] = cur[L0 + k]; tsum += cnt8[k]; }
  unsigned winclU = waveInclScanU(tsum, lane);
  if (lane == 31) redU[wid] = winclU;
  __syncthreads();
  if (wid == 0) {
    unsigned wv = redU[lane];
    unsigned wi = waveInclScanU(wv, lane);
    redU[lane] = wi - wv;                                  // exclusive wave base
  }
  __syncthreads();
  unsigned start8[SEG];
  {
    unsigned run = redU[wid] + (winclU - tsum);
#pragma unroll
    for (int k = 0; k < SEG; ++k) { start8[k] = run; run += cnt8[k]; }
  }
#pragma unroll
  for (int k = 0; k < SEG; ++k) cur[L0 + k] = start8[k];   // cursors
  __syncthreads();

  // ---- P3: scatter original indices into buckets (ds_add_rtn) ----
#pragma unroll
  for (int k = 0; k < SEG; ++k) {
    int i = tid + k * BLOCK;
    unsigned pos = atomicAdd(&cur[lab[i]], 1u);
    sidx[pos] = (unsigned short)i;
  }
  __syncthreads();

  // ---- P4: per-bucket stable sort (tiny) + deterministic per-label exp sums ----
  float es[SEG];
  float tes = 0.0f;
#pragma unroll
  for (int k = 0; k < SEG; ++k) {
    unsigned st = start8[k];
    unsigned en = st + cnt8[k];
    for (unsigned a = st + 1; a < en; ++a) {               // insertion sort by index
      unsigned short key = sidx[a];
      unsigned p = a;
      while (p > st && sidx[p - 1] > key) { sidx[p] = sidx[p - 1]; --p; }
      sidx[p] = key;
    }
    float e = 0.0f;
    for (unsigned a = st; a < en; ++a) e += vexp[sidx[a]];
    es[k] = e;
    tes += e;
  }

  // block inclusive scan over per-label exp sums (for suffix sums G[L])
  float winclF = waveInclScanF(tes, lane);
  if (lane == 31) redF[wid] = winclF;
  __syncthreads();
  if (wid == 0) {
    float wv = redF[lane];
    float wi = waveInclScanF(wv, lane);
    redF[lane] = wi - wv;
    if (lane == 31) redF[32] = wi;                         // grand total of exp
  }
  __syncthreads();
  float baseE  = redF[wid] + (winclF - tes);               // exclusive prefix before L0
  float totalE = redF[32];

  // ---- P5: reverse cumulative logsumexp within each bucket ----
  float accL = 0.0f;
  {
    float incl = baseE;
#pragma unroll
    for (int k = 0; k < SEG; ++k) {
      incl += es[k];
      float g = totalE - incl;                             // sum of exp over labels > L
      unsigned st = start8[k];
      int n = (int)cnt8[k];
      float runG = g;
      for (int m = n - 1; m >= 0; --m) {
        runG += vexp[sidx[st + m]];
        accL += logf(runG);
      }
    }
  }

  // ---- P6: row reduction: sum(lse) - sum(s) ----
  float v = accL - ssum;
  __syncthreads();                                         // redF reuse guard
#pragma unroll
  for (int d = 16; d > 0; d >>= 1) v += __shfl_down(v, d, 32);
  if (lane == 0) redF[wid] = v;
  __syncthreads();
  if (wid == 0) {
    float x = redF[lane];
#pragma unroll
    for (int d = 16; d > 0; d >>= 1) x += __shfl_down(x, d, 32);
    if (lane == 0) row_out[row] = x;
  }
}

__global__ __launch_bounds__(BLOCK) void pl_finalize_kernel(const float* __restrict__ row_out,
                                                            float* __restrict__ out, int B) {
  __shared__ float red[32];
  const int tid = threadIdx.x, lane = tid & 31, wid = tid >> 5;
  float v = 0.0f;
  for (int i = tid; i < B; i += BLOCK) v += row_out[i];
#pragma unroll
  for (int d = 16; d > 0; d >>= 1) v += __shfl_down(v, d, 32);
  if (lane == 0) red[wid] = v;
  __syncthreads();
  if (wid == 0) {
    float x = red[lane];
#pragma unroll
    for (int d = 16; d > 0; d >>= 1) x += __shfl_down(x, d, 32);
    if (lane == 0) out[0] = (float)((double)x / ((double)B * (double)T_LEN));
  }
}

extern "C" void kernel_launch(void* const* d_in, const int* in_sizes, int n_in,
                              void* d_out, int out_size, void* d_ws, size_t ws_size,
                              hipStream_t stream) {
  const float* logits = (const float*)d_in[0];
  const int*   labels = (const int*)d_in[1];
  float* out  = (float*)d_out;
  float* rows = (float*)d_ws;                 // B floats of scratch, overwritten each call
  const int B = in_sizes[0] / T_LEN;

  pl_row_kernel<<<B, BLOCK, SMEM_BYTES, stream>>>(logits, labels, rows);
  pl_finalize_kernel<<<1, BLOCK, 0, stream>>>(rows, out, B);
}